// VirtualNodePyg_90718299226161
// MI455X (gfx1250) — compile-verified
//
#include <hip/hip_runtime.h>

typedef float v2f __attribute__((ext_vector_type(2)));
typedef float v8f __attribute__((ext_vector_type(8)));

#define POOL_CHUNK 64

// --- zero the pooling accumulator -------------------------------------------
__global__ void vn_zero_kernel(float* __restrict__ p, int n) {
  int i = blockIdx.x * blockDim.x + threadIdx.x;
  if (i < n) p[i] = 0.0f;
}

// --- segment-sum pooling over sorted batch ids ------------------------------
// blockDim.x == D/2 ; each thread owns a float2 column pair; one block walks
// POOL_CHUNK consecutive node rows, accumulating in registers and flushing to
// pool[] with atomics only when the (sorted) segment id changes.
__global__ void vn_pool_kernel(const float* __restrict__ h,
                               const int* __restrict__ batch,
                               float* __restrict__ pool,
                               int N, int D) {
  const int c = threadIdx.x;                 // column-pair index
  const int row0 = blockIdx.x * POOL_CHUNK;
  const int row1 = min(row0 + POOL_CHUNK, N);
  float ax = 0.0f, ay = 0.0f;
  int seg = batch[row0];
  for (int r = row0; r < row1; ++r) {
    const int s = batch[r];
    if (s != seg) {
      atomicAdd(&pool[(size_t)seg * D + 2 * c], ax);
      atomicAdd(&pool[(size_t)seg * D + 2 * c + 1], ay);
      ax = 0.0f; ay = 0.0f;
      seg = s;
    }
    const float2 hv = *(const float2*)(h + (size_t)r * D + 2 * c);
    ax += hv.x; ay += hv.y;
  }
  atomicAdd(&pool[(size_t)seg * D + 2 * c], ax);
  atomicAdd(&pool[(size_t)seg * D + 2 * c + 1], ay);
}

// --- fused FC layer with f32 WMMA -------------------------------------------
// vn_out = vn_h + relu((vn_h + pool) @ W^T + bias)
// One wave (32 lanes) computes one 16x16 output tile; K stepped by 4 via
// V_WMMA_F32_16X16X4_F32 (full f32 precision; GEMM cost is negligible).
// A 16x4 f32 layout: lanes 0-15 M=0..15 {K=0,K=1}, lanes 16-31 M=0..15 {K=2,K=3}.
// B 4x16 mirrors: N = lane&15, K pair = 2*(lane>>4). B[k][n] = W[n][k].
__global__ void __launch_bounds__(32)
vn_fc_wmma_kernel(const float* __restrict__ vn_h,
                  const float* __restrict__ pool,
                  const float* __restrict__ W,
                  const float* __restrict__ bias,
                  float* __restrict__ vn_out,
                  int D) {
  const int lane = threadIdx.x;
  const int half = lane >> 4;          // 0 or 1
  const int l16  = lane & 15;
  const int n0 = blockIdx.x * 16;      // output column tile
  const int m0 = blockIdx.y * 16;      // output row tile (graphs)
  const size_t arow = (size_t)(m0 + l16) * D;   // A row for this lane
  const size_t brow = (size_t)(n0 + l16) * D;   // W row = B column

  v8f c = {};
  for (int k = 0; k < D; k += 4) {
    const int ka = k + 2 * half;
    const float2 va = *(const float2*)(vn_h + arow + ka);
    const float2 vp = *(const float2*)(pool + arow + ka);
    const float2 vb = *(const float2*)(W + brow + ka);
    v2f a, b;
    a.x = va.x + vp.x; a.y = va.y + vp.y;
    b.x = vb.x;        b.y = vb.y;
    // 8 args: (neg_a, A, neg_b, B, c_mod, C, reuse_a, reuse_b)
    c = __builtin_amdgcn_wmma_f32_16x16x4_f32(
        false, a, false, b, (short)0, c, false, false);
  }

  // C/D layout: VGPR r -> M = r + 8*half (lanes>=16), N = lane&15.
#pragma unroll
  for (int r = 0; r < 8; ++r) {
    const int row = m0 + r + 8 * half;
    const int col = n0 + l16;
    float v = c[r] + bias[col];
    v = v > 0.0f ? v : 0.0f;                       // relu
    vn_out[(size_t)row * D + col] = vn_h[(size_t)row * D + col] + v;  // residual
  }
}

// --- broadcast-gather with residual: h_new = h + vn_new[batch] --------------
// One block per node row; blockDim.x == D/4, float4 per thread (full 2KB row).
__global__ void vn_gather_kernel(const float* __restrict__ h,
                                 const int* __restrict__ batch,
                                 const float* __restrict__ vn_new,
                                 float* __restrict__ h_out,
                                 int D) {
  const int row = blockIdx.x;
  const int t = threadIdx.x;
  const int g = batch[row];
  const size_t hoff = (size_t)row * D + 4 * t;
  const float4 hv = *(const float4*)(h + hoff);
  const float4 vv = *(const float4*)(vn_new + (size_t)g * D + 4 * t);
  float4 o;
  o.x = hv.x + vv.x; o.y = hv.y + vv.y;
  o.z = hv.z + vv.z; o.w = hv.w + vv.w;
  *(float4*)(h_out + hoff) = o;
}

extern "C" void kernel_launch(void* const* d_in, const int* in_sizes, int n_in,
                              void* d_out, int out_size, void* d_ws, size_t ws_size,
                              hipStream_t stream) {
  const float* h     = (const float*)d_in[0];
  const float* vn_h  = (const float*)d_in[1];
  const int*   batch = (const int*)d_in[2];
  const float* W     = (const float*)d_in[4];
  const float* bias  = (const float*)d_in[5];

  const int D  = in_sizes[5];          // feature dim (512)
  const int Bg = in_sizes[1] / D;      // num graphs (256)
  const int N  = in_sizes[2];          // num nodes (200000)

  float* h_out  = (float*)d_out;                       // [N, D]
  float* vn_out = (float*)d_out + (size_t)N * D;       // [B, D]
  float* pool   = (float*)d_ws;                        // [B, D] scratch

  // 1) zero pool accumulator
  {
    const int n = Bg * D;
    vn_zero_kernel<<<(n + 255) / 256, 256, 0, stream>>>(pool, n);
  }
  // 2) segment-sum pooling (register accumulation, boundary atomics only)
  vn_pool_kernel<<<(N + POOL_CHUNK - 1) / POOL_CHUNK, D / 2, 0, stream>>>(
      h, batch, pool, N, D);
  // 3) fused FC + relu + residual on virtual-node state, f32 WMMA
  vn_fc_wmma_kernel<<<dim3(D / 16, Bg / 16), 32, 0, stream>>>(
      vn_h, pool, W, bias, vn_out, D);
  // 4) h_new = h + vn_h_new[batch]
  vn_gather_kernel<<<N, D / 4, 0, stream>>>(h, batch, vn_out, h_out, D);
}